// SimpleGCNConv_28389733827054
// MI455X (gfx1250) — compile-verified
//
#include <hip/hip_runtime.h>

#define N_NODES 8192
#define C 128            // IN_C == OUT_C
#define KC 32            // K-chunk per LDS stage
#define PAD 36           // LDS row stride (dwords): 16B-aligned rows, conflict-free
#define ROWS_PER_BLOCK 128
#define CHUNK_A (ROWS_PER_BLOCK * PAD)   // dwords per adj panel buffer
#define CHUNK_X (C * PAD)                // dwords per xsT panel buffer

typedef __attribute__((ext_vector_type(2))) float v2f;
typedef __attribute__((ext_vector_type(4))) float v4f;
typedef __attribute__((ext_vector_type(8))) float v8f;
typedef __attribute__((ext_vector_type(4))) unsigned int u32x4;
typedef __attribute__((ext_vector_type(8))) int i32x8;
typedef __attribute__((ext_vector_type(4))) int i32x4;

// ---------------------------------------------------------------------------
// Kernel 1: deg = rowsum(adj) + 1 ; dinv = deg^-1/2 ; xs = dinv[:,None] * x
// ---------------------------------------------------------------------------
__global__ __launch_bounds__(256) void gcn_deg_scale(const float* __restrict__ adj,
                                                     const float* __restrict__ x,
                                                     float* __restrict__ dinv,
                                                     float* __restrict__ xs) {
  const int row = blockIdx.x;
  const v4f* a4 = (const v4f*)(adj + (size_t)row * N_NODES);
  float s = 0.f;
  for (int i = threadIdx.x; i < N_NODES / 4; i += 256) {
    v4f v = a4[i];
    s += (v.x + v.y) + (v.z + v.w);
  }
  #pragma unroll
  for (int off = 16; off > 0; off >>= 1)
    s += __shfl_down(s, off, 32);
  __shared__ float red[8];
  const int wave = threadIdx.x >> 5;
  const int lane = threadIdx.x & 31;
  if (lane == 0) red[wave] = s;
  __syncthreads();
  float deg = 1.0f;  // self loop
  #pragma unroll
  for (int w = 0; w < 8; ++w) deg += red[w];
  const float di = (deg > 0.f) ? rsqrtf(deg) : 0.f;
  if (threadIdx.x == 0) dinv[row] = di;
  if (threadIdx.x < C)
    xs[(size_t)row * C + threadIdx.x] = di * x[(size_t)row * C + threadIdx.x];
}

// ---------------------------------------------------------------------------
// TDM: DMA one adj panel (128 rows x KC fp32, row stride N_NODES) into LDS,
// with hardware padding (+4 dwords every 32 dwords) to produce PAD=36 rows.
// D# layout per CDNA5 ISA ch.8 (group0 128b, group1 256b; groups 2/3 unused).
// This toolchain's builtin takes 6 args (g0, g1, g2, g3, extra group, cpol).
// ---------------------------------------------------------------------------
#if __has_builtin(__builtin_amdgcn_tensor_load_to_lds)
#define HAVE_TDM 1
__device__ __forceinline__ void tdm_load_adj_panel(const float* gsrc, unsigned lds_byte) {
  const unsigned long long ga = (unsigned long long)(uintptr_t)gsrc;
  u32x4 g0;
  g0[0] = 1u;                                      // count=1, user descriptor
  g0[1] = lds_byte;                                // lds_addr (bytes, wave-relative)
  g0[2] = (unsigned)ga;                            // global_addr[31:0]
  g0[3] = (unsigned)((ga >> 32) & 0x1FFFFFFu)      // global_addr[56:32]
          | (2u << 30);                            // type = 2 ("image")
  i32x8 g1;
  g1[0] = (2 << 16)     // data_size = 4 bytes
        | (1 << 20)     // pad_enable
        | (4 << 22)     // pad_interval: code 4 -> every 32 dwords
        | (3 << 25);    // pad_amount:   code 3 -> 4 dwords  (32+4 = PAD)
  g1[1] = (int)((8192u & 0xFFFFu) << 16);          // tensor_dim0[15:0] = 8192
  g1[2] = (int)((8192u >> 16) | ((8192u & 0xFFFFu) << 16)); // dim0 hi | dim1 lo
  g1[3] = (int)((8192u >> 16) | ((unsigned)KC << 16));      // dim1 hi | tile_dim0=KC
  g1[4] = (int)ROWS_PER_BLOCK;                     // tile_dim1 = 128, tile_dim2 = 0
  g1[5] = (int)N_NODES;                            // tensor_dim0_stride[31:0]
  g1[6] = 0;                                       // stride hi | dim1_stride lo
  g1[7] = 0;
  i32x4 z4 = {0, 0, 0, 0};
  i32x8 z8 = {0, 0, 0, 0, 0, 0, 0, 0};
  __builtin_amdgcn_tensor_load_to_lds(g0, g1, z4, z4, z8, 0);
}
#else
#define HAVE_TDM 0
#endif

// ---------------------------------------------------------------------------
// Kernel 2: support = dinv[:,None] * (adj @ xs + xs)
// 64 blocks x 8 waves; TDM double-buffers the adj panel while waves run WMMA.
// ---------------------------------------------------------------------------
__global__ __launch_bounds__(256) void gcn_aggregate(const float* __restrict__ adj,
                                                     const float* __restrict__ xs,
                                                     const float* __restrict__ dinv,
                                                     float* __restrict__ support) {
  __shared__ float sA[2 * CHUNK_A];  // adj panels (double buffered)
  __shared__ float sX[2 * CHUNK_X];  // xsT panels (double buffered)

  const int tid  = threadIdx.x;
  const int wave = tid >> 5;
  const int lane = tid & 31;
  const int half = lane >> 4;
  const int l16  = lane & 15;
  const int rowBase = blockIdx.x * ROWS_PER_BLOCK;

  v8f acc[8];
  {
    v8f z = {};
    #pragma unroll
    for (int t = 0; t < 8; ++t) acc[t] = z;
  }

  // generic->LDS: low 32 bits of the generic address are the LDS offset
  const unsigned ldsA0 = (unsigned)(uintptr_t)(void*)&sA[0];

  // ---- stage helpers -----------------------------------------------------
  auto stage_adj = [&](int k0, float* dstA) {
#if HAVE_TDM
    if (wave == 0)
      tdm_load_adj_panel(adj + (size_t)rowBase * N_NODES + k0,
                         ldsA0 + (unsigned)((dstA - &sA[0]) * 4));
#else
    const int r = tid >> 3, q = tid & 7;
    #pragma unroll
    for (int it = 0; it < 4; ++it) {
      const int rr = r + it * 32;
      v4f v = *(const v4f*)(adj + (size_t)(rowBase + rr) * N_NODES + k0 + q * 4);
      float* dst = &dstA[rr * PAD + q * 4];
      dst[0] = v.x; dst[1] = v.y; dst[2] = v.z; dst[3] = v.w;
    }
#endif
  };
  auto stage_xsT = [&](int k0, float* dstX) {
    #pragma unroll
    for (int it = 0; it < 4; ++it) {
      const int idx = it * 256 + tid;   // 1024 float4 units
      const int kr  = idx >> 5;         // 0..31
      const int q   = idx & 31;         // col/4
      v4f v = *(const v4f*)(xs + (size_t)(k0 + kr) * C + q * 4);
      dstX[(q * 4 + 0) * PAD + kr] = v.x;
      dstX[(q * 4 + 1) * PAD + kr] = v.y;
      dstX[(q * 4 + 2) * PAD + kr] = v.z;
      dstX[(q * 4 + 3) * PAD + kr] = v.w;
    }
  };

  // ---- prologue: fill buffer 0 ------------------------------------------
  stage_adj(0, &sA[0]);
  stage_xsT(0, &sX[0]);
#if HAVE_TDM
  if (wave == 0) __builtin_amdgcn_s_wait_tensorcnt(0);
#endif
  __syncthreads();

  int cur = 0;
  const int aRow = wave * 16 + l16;
  for (int k0 = 0; k0 < N_NODES; k0 += KC) {
    const int nxt = cur ^ 1;
    // overlap: DMA/stage next panels while computing on current
    if (k0 + KC < N_NODES) {
      stage_adj(k0 + KC, &sA[nxt * CHUNK_A]);
      stage_xsT(k0 + KC, &sX[nxt * CHUNK_X]);
    }

    const float* bufA = &sA[cur * CHUNK_A];
    const float* bufX = &sX[cur * CHUNK_X];
    #pragma unroll
    for (int kk = 0; kk < KC; kk += 4) {
      // A 16x4: lanes 0-15 -> K {kk,kk+1}; lanes 16-31 -> K {kk+2,kk+3}
      v2f a = *(const v2f*)&bufA[aRow * PAD + kk + half * 2];
      #pragma unroll
      for (int t = 0; t < 8; ++t) {
        v2f b = *(const v2f*)&bufX[(t * 16 + l16) * PAD + kk + half * 2];
        acc[t] = __builtin_amdgcn_wmma_f32_16x16x4_f32(
            false, a, false, b, (short)0, acc[t], false, false);
      }
    }

#if HAVE_TDM
    if (wave == 0) __builtin_amdgcn_s_wait_tensorcnt(0);
#endif
    __syncthreads();
    cur = nxt;
  }

  // ---- epilogue: support[r] = dinv[r] * (acc + xs[r]) (self-loop fused) --
  const int r0 = rowBase + wave * 16;
  #pragma unroll
  for (int v = 0; v < 8; ++v) {
    const int r = r0 + v + half * 8;   // C/D layout: VGPR v -> M=v (lo) / v+8 (hi)
    const float di = dinv[r];
    #pragma unroll
    for (int t = 0; t < 8; ++t) {
      const int col = t * 16 + l16;
      support[(size_t)r * C + col] = di * (acc[t][v] + xs[(size_t)r * C + col]);
    }
  }
}

// ---------------------------------------------------------------------------
// Kernel 3: out = support @ W.T + b   (cache-resident, tiny)
// ---------------------------------------------------------------------------
__global__ __launch_bounds__(256) void gcn_linear(const float* __restrict__ support,
                                                  const float* __restrict__ W,
                                                  const float* __restrict__ bias,
                                                  float* __restrict__ out) {
  const int tid  = threadIdx.x;
  const int wave = tid >> 5;
  const int lane = tid & 31;
  const int half = lane >> 4;
  const int l16  = lane & 15;
  const int r0 = blockIdx.x * ROWS_PER_BLOCK + wave * 16;

  v8f acc[8];
  {
    v8f z = {};
    #pragma unroll
    for (int t = 0; t < 8; ++t) acc[t] = z;
  }

  #pragma unroll 4
  for (int c0 = 0; c0 < C; c0 += 4) {
    v2f a = *(const v2f*)(support + (size_t)(r0 + l16) * C + c0 + half * 2);
    #pragma unroll
    for (int t = 0; t < 8; ++t) {
      v2f b = *(const v2f*)(W + (size_t)(t * 16 + l16) * C + c0 + half * 2);
      acc[t] = __builtin_amdgcn_wmma_f32_16x16x4_f32(
          false, a, false, b, (short)0, acc[t], false, false);
    }
  }

  #pragma unroll
  for (int v = 0; v < 8; ++v) {
    const int r = r0 + v + half * 8;
    #pragma unroll
    for (int t = 0; t < 8; ++t) {
      const int col = t * 16 + l16;
      out[(size_t)r * C + col] = acc[t][v] + bias[col];
    }
  }
}

// ---------------------------------------------------------------------------
extern "C" void kernel_launch(void* const* d_in, const int* in_sizes, int n_in,
                              void* d_out, int out_size, void* d_ws, size_t ws_size,
                              hipStream_t stream) {
  (void)in_sizes; (void)n_in; (void)out_size; (void)ws_size;
  const float* x   = (const float*)d_in[0];  // (8192, 128)
  const float* adj = (const float*)d_in[1];  // (8192, 8192)
  const float* W   = (const float*)d_in[2];  // (128, 128)
  const float* b   = (const float*)d_in[3];  // (128,)
  float* out = (float*)d_out;                // (8192, 128)

  float* dinv    = (float*)d_ws;
  float* xs      = dinv + N_NODES;
  float* support = xs + (size_t)N_NODES * C;

  gcn_deg_scale<<<N_NODES, 256, 0, stream>>>(adj, x, dinv, xs);
  gcn_aggregate<<<N_NODES / ROWS_PER_BLOCK, 256, 0, stream>>>(adj, xs, dinv, support);
  gcn_linear<<<N_NODES / ROWS_PER_BLOCK, 256, 0, stream>>>(support, W, b, out);
}